// SelfAttention_927712936365
// MI455X (gfx1250) — compile-verified
//
#include <hip/hip_runtime.h>
#include <math.h>

#define B_  32
#define S_  4096
#define U_  512

typedef float v2f __attribute__((ext_vector_type(2)));
typedef float v8f __attribute__((ext_vector_type(8)));

// ---------------------------------------------------------------------------
// K1: uv[u] = sum_v U_w[u,v]*V_w[v];  wv[u] = sum_v W_w[u,v]*V_w[v]
// ---------------------------------------------------------------------------
__global__ void k_uvwv(const float* __restrict__ Uw, const float* __restrict__ Ww,
                       const float* __restrict__ Vw,
                       float* __restrict__ uv, float* __restrict__ wv) {
    int u = blockIdx.x * blockDim.x + threadIdx.x;
    if (u < U_) {
        float a = 0.f, b = 0.f;
        for (int v = 0; v < U_; ++v) {
            float vw = Vw[v];
            a += Uw[u * U_ + v] * vw;
            b += Ww[u * U_ + v] * vw;
        }
        uv[u] = a;
        wv[u] = b;
    }
}

// ---------------------------------------------------------------------------
// K2: c[b] = s_prev[b]·wv + (W_b + U_b)·V_w + V_b
// ---------------------------------------------------------------------------
__global__ void k_cb(const float* __restrict__ sprev, const float* __restrict__ wv,
                     const float* __restrict__ Wb, const float* __restrict__ Ub,
                     const float* __restrict__ Vw, const float* __restrict__ Vb,
                     float* __restrict__ cb) {
    __shared__ float red[64];
    int b = blockIdx.x, t = threadIdx.x;
    float acc = 0.f;
    for (int u = t; u < U_; u += 64)
        acc += sprev[b * U_ + u] * wv[u] + (Wb[u] + Ub[u]) * Vw[u];
    red[t] = acc;
    __syncthreads();
    for (int off = 32; off > 0; off >>= 1) {
        if (t < off) red[t] += red[t + off];
        __syncthreads();
    }
    if (t == 0) cb[b] = red[0] + Vb[0];
}

// ---------------------------------------------------------------------------
// K3: e[b,s] = tanh( h[b,s]·uv + c[b] )  via V_WMMA_F32_16X16X4_F32.
// One wave per 16 s-rows: A = 16x4 fp32 tile of h (rows striped over lanes,
// lane-half selects K in {0,2}); B = uv[k..k+3] broadcast to all 16 columns,
// so every output column of C holds the same 16 dot products. Memory-bound
// (streams the 256 MB hidden tensor), so the wasted N columns are free.
// ---------------------------------------------------------------------------
__global__ void k_logits(const float* __restrict__ h, const float* __restrict__ uv,
                         const float* __restrict__ cb, float* __restrict__ e) {
    __shared__ float suv[U_];
    int t = threadIdx.x;
    for (int i = t; i < U_; i += 256) suv[i] = uv[i];
    __syncthreads();

    int wave = t >> 5;
    int lane = t & 31;
    int tile = blockIdx.x * 8 + wave;          // 8192 tiles of 16 rows = B*S rows
    int b    = tile >> 8;                      // 256 tiles per batch (4096/16)
    int srow = (tile & 255) * 16;

    const float* hb = h + ((size_t)b * S_ + srow) * U_;
    int m    = lane & 15;                      // A row owned by this lane
    int koff = (lane >> 4) * 2;                // lanes 0-15: K=0,1 ; 16-31: K=2,3

    v8f c = {};
    for (int k = 0; k < U_; k += 4) {
        if ((k & 63) == 0)                     // stream-ahead hint -> global_prefetch_b8
            __builtin_prefetch(hb + (size_t)m * U_ + k + 128, 0, 3);
        v2f a  = *(const v2f*)(hb + (size_t)m * U_ + k + koff);   // global_load_b64
        v2f bb = *(const v2f*)(&suv[k + koff]);                   // ds_load_b64
        c = __builtin_amdgcn_wmma_f32_16x16x4_f32(
                /*neg_a=*/false, a, /*neg_b=*/false, bb,
                /*c_mod=*/(short)0, c, /*reuse_a=*/false, /*reuse_b=*/false);
    }

    float cbb = cb[b];
    // C layout: lanes 0-15 hold M=vgpr (rows 0-7), lanes 16-31 hold M=8+vgpr.
    // All columns identical -> lanes {0-7} emit rows 0-7, lanes {16-23} rows 8-15.
    union { v8f v; float f[8]; } uu;
    uu.v = c;
    int   l7  = lane & 7;
    float val = uu.f[l7];
    int   row = l7 + ((lane & 16) ? 8 : 0);
    if ((lane & 8) == 0)
        e[(size_t)b * S_ + srow + row] = tanhf(val + cbb);
}

// ---------------------------------------------------------------------------
// K4: in-place softmax over S for each batch (max-subtracted, like jax.nn.softmax)
// ---------------------------------------------------------------------------
__global__ void k_softmax(float* __restrict__ w) {
    __shared__ float red[256];
    int b = blockIdx.x, t = threadIdx.x;
    float* wb = w + (size_t)b * S_;

    float mx = -1e30f;
    for (int s = t; s < S_; s += 256) mx = fmaxf(mx, wb[s]);
    red[t] = mx;
    __syncthreads();
    for (int off = 128; off > 0; off >>= 1) {
        if (t < off) red[t] = fmaxf(red[t], red[t + off]);
        __syncthreads();
    }
    mx = red[0];
    __syncthreads();

    float sum = 0.f;
    for (int s = t; s < S_; s += 256) sum += __expf(wb[s] - mx);
    red[t] = sum;
    __syncthreads();
    for (int off = 128; off > 0; off >>= 1) {
        if (t < off) red[t] += red[t + off];
        __syncthreads();
    }
    float inv = 1.0f / red[0];
    for (int s = t; s < S_; s += 256) wb[s] = __expf(wb[s] - mx) * inv;
}

// ---------------------------------------------------------------------------
// K5a: partial context over an S-chunk (16 chunks -> 512 blocks to fill HBM BW)
// K5b: deterministic reduce of the 16 partials
// ---------------------------------------------------------------------------
#define NSPLIT 16
__global__ void k_part(const float* __restrict__ h, const float* __restrict__ w,
                       float* __restrict__ part) {
    int b = blockIdx.x, p = blockIdx.y, t = threadIdx.x;
    const int chunk = S_ / NSPLIT;             // 256
    const float* hb = h + (size_t)b * S_ * U_;
    const float* wb = w + (size_t)b * S_;
    float a0 = 0.f, a1 = 0.f;
    int s0 = p * chunk;
    for (int s = s0; s < s0 + chunk; ++s) {
        float wv = wb[s];
        a0 += wv * hb[(size_t)s * U_ + t];
        a1 += wv * hb[(size_t)s * U_ + t + 256];
    }
    float* pp = part + ((size_t)p * B_ + b) * U_;
    pp[t]       = a0;
    pp[t + 256] = a1;
}

__global__ void k_reduce(const float* __restrict__ part, float* __restrict__ ctx) {
    int b = blockIdx.x, t = threadIdx.x;
    float a0 = 0.f, a1 = 0.f;
    for (int p = 0; p < NSPLIT; ++p) {
        const float* pp = part + ((size_t)p * B_ + b) * U_;
        a0 += pp[t];
        a1 += pp[t + 256];
    }
    ctx[(size_t)b * U_ + t]       = a0;
    ctx[(size_t)b * U_ + t + 256] = a1;
}

// ---------------------------------------------------------------------------
// launch
// ---------------------------------------------------------------------------
extern "C" void kernel_launch(void* const* d_in, const int* in_sizes, int n_in,
                              void* d_out, int out_size, void* d_ws, size_t ws_size,
                              hipStream_t stream) {
    const float* s_prev = (const float*)d_in[0];   // [32,512]
    const float* hidden = (const float*)d_in[1];   // [32,4096,512]
    const float* W_w    = (const float*)d_in[2];   // [512,512]
    const float* W_b    = (const float*)d_in[3];   // [512]
    const float* U_w    = (const float*)d_in[4];   // [512,512]
    const float* U_b    = (const float*)d_in[5];   // [512]
    const float* V_w    = (const float*)d_in[6];   // [512,1]
    const float* V_b    = (const float*)d_in[7];   // [1]

    float* out     = (float*)d_out;
    float* context = out;                 // [32,512]
    float* weights = out + B_ * U_;       // [32,4096]  (logits, then softmax in place)

    float* ws   = (float*)d_ws;
    float* uv   = ws;                     // 512
    float* wv   = ws + 512;               // 512
    float* cb   = ws + 1024;              // 32
    float* part = ws + 2048;              // NSPLIT*32*512 = 256K floats

    k_uvwv  <<<2, 256, 0, stream>>>(U_w, W_w, V_w, uv, wv);
    k_cb    <<<B_, 64, 0, stream>>>(s_prev, wv, W_b, U_b, V_w, V_b, cb);
    k_logits<<<(B_ * S_) / 128, 256, 0, stream>>>(hidden, uv, cb, weights);
    k_softmax<<<B_, 256, 0, stream>>>(weights);
    k_part  <<<dim3(B_, NSPLIT), 256, 0, stream>>>(hidden, weights, part);
    k_reduce<<<B_, 256, 0, stream>>>(part, context);
}